// EuESN_maml_42657615184236
// MI455X (gfx1250) — compile-verified
//
#include <hip/hip_runtime.h>

// ---------------------------------------------------------------------------
// Wave-equation transition-matrix assembly (3N x 3N sparse scatter, N = 48^2).
// Memory-bound: ~191 MB of stores, ~8.2 us floor at 23.3 TB/s HBM.
// Zero-fill uses gfx1250 async LDS->global stores (ASYNCcnt path).
// ---------------------------------------------------------------------------

#define GRID_N 48
#define NN     (GRID_N * GRID_N)   // 2304
#define MM     (3 * NN)            // 6912

typedef float f32x4 __attribute__((ext_vector_type(4)));
typedef int   i32x4 __attribute__((ext_vector_type(4)));

#if defined(__gfx1250__) && __has_builtin(__builtin_amdgcn_global_store_async_from_lds_b128)
#define USE_ASYNC_LDS_STORE 1
#define AS_GLOBAL __attribute__((address_space(1)))
#define AS_LOCAL  __attribute__((address_space(3)))
#else
#define USE_ASYNC_LDS_STORE 0
#endif

// ---- kernel 1: dx = (DT/CN) * max(c) * sqrt(2), f32 op order as in JAX ----
__global__ void maxdx_kernel(const float* __restrict__ c, float* __restrict__ ws) {
    __shared__ float red[256];
    float m = -3.0e38f;
    for (int i = threadIdx.x; i < NN; i += 256) m = fmaxf(m, c[i]);
    red[threadIdx.x] = m;
    __syncthreads();
    for (int s = 128; s > 0; s >>= 1) {
        if (threadIdx.x < s) red[threadIdx.x] = fmaxf(red[threadIdx.x], red[threadIdx.x + s]);
        __syncthreads();
    }
    if (threadIdx.x == 0) {
        float dx = 10.0f * red[0];      // DT/CN = 10 exactly
        dx = dx * 1.41421356237309515f; // * sqrt(2), rounded to f32
        ws[0] = dx;
    }
}

// ---- kernel 2: zero-fill the whole 6912x6912 output with b128 stores ------
__global__ void wave_zero_kernel(float* __restrict__ out, long long n4) {
    long long tid    = (long long)blockIdx.x * blockDim.x + threadIdx.x;
    long long stride = (long long)gridDim.x * blockDim.x;
    f32x4* o4 = (f32x4*)out;
#if USE_ASYNC_LDS_STORE
    // Stream zeros from a zeroed LDS staging tile via the CDNA5 async path
    // (ASYNCcnt-tracked, no VGPR store data in flight).
    __shared__ __align__(16) float zbuf[256 * 4];
    ((f32x4*)zbuf)[threadIdx.x] = (f32x4)0.0f;
    __syncthreads();
    AS_LOCAL i32x4* lsrc = (AS_LOCAL i32x4*)&((f32x4*)zbuf)[threadIdx.x];
    for (long long i = tid; i < n4; i += stride) {
        __builtin_amdgcn_global_store_async_from_lds_b128(
            (AS_GLOBAL i32x4*)(o4 + i), lsrc, /*offset=*/0, /*cpol=*/0);
    }
    asm volatile("s_wait_asynccnt 0" ::: "memory");
#else
    f32x4 z = (f32x4)0.0f;
    for (long long i = tid; i < n4; i += stride)
        __builtin_nontemporal_store(z, o4 + i);  // NT: 191 MB stream, no reuse
#endif
}

// ---- kernel 3: scatter the <=11 nonzeros owned by lattice point k ---------
// Flattening matches the reference: idx k = col*n + row over the n x n grid,
// with inputs stored row-major [row*n + col].
__global__ void scatter_kernel(const float* __restrict__ c,
                               const float* __restrict__ dkx,
                               const float* __restrict__ dky,
                               const float* __restrict__ ws,
                               float* __restrict__ A) {
    int k = blockIdx.x * blockDim.x + threadIdx.x;
    if (k >= NN) return;

    const int col = k / GRID_N;      // block index (i-1)
    const int row = k % GRID_N;      // within-block index (ii-1)
    const int src = row * GRID_N + col;

    const float dx = ws[0];
    const float rv = c[src] / dx;                       // rvec[k]
    const float gx = dkx[src] * (float)col * 0.5f;      // gx[k]
    const float gy = dky[src] * (float)col * 0.5f;      // gy[k]

    const float diagM  = 0.99995f;   // 1/DT - KP/2
    const float diagMp = 1.00005f;   // 1/DT + KP/2
    const float mi  = 1.0f / diagM;
    const float nxi = 1.0f / (1.0f + gx);
    const float nyi = 1.0f / (1.0f + gy);

    // --- row block 0: [A00 | A01 | A02], global row = k ---
    {
        size_t r0 = (size_t)k * MM;
        A[r0 + k]               = mi * diagMp;          // A00 diag
        A[r0 + NN + k]          = mi * (-rv);           // A01 diag  (mi * Sx)
        if (k >= GRID_N)
            A[r0 + NN + k - GRID_N] = mi * rv;          // A01 sub-n diag
        A[r0 + 2 * NN + k]      = mi * (-rv);           // A02 diag  (mi * Sy)
        if (row != 0)
            A[r0 + 2 * NN + k - 1]  = mi * rv;          // A02 sub-1 diag (bmask)
    }
    // --- row block 1: [A10 | A11 | 0], global row = NN + k ---
    {
        size_t r1 = (size_t)(NN + k) * MM;
        A[r1 + k]               = nxi * rv;             // A10 diag  (nxi * Rx)
        if (k < NN - GRID_N)
            A[r1 + k + GRID_N]  = nxi * (-rv);          // A10 super-n diag
        A[r1 + NN + k]          = nxi * (1.0f - gx);    // A11 diag
    }
    // --- row block 2: [A20 | 0 | A22], global row = 2NN + k ---
    {
        size_t r2 = (size_t)(2 * NN + k) * MM;
        A[r2 + k]               = nyi * rv;             // A20 diag  (nyi * Ry)
        if (row != GRID_N - 1)
            A[r2 + k + 1]       = nyi * (-rv);          // A20 super-1 diag (bmask)
        A[r2 + 2 * NN + k]      = nyi * (1.0f - gy);    // A22 diag
    }
}

extern "C" void kernel_launch(void* const* d_in, const int* in_sizes, int n_in,
                              void* d_out, int out_size, void* d_ws, size_t ws_size,
                              hipStream_t stream) {
    (void)in_sizes; (void)n_in; (void)ws_size;
    const float* c   = (const float*)d_in[0];
    const float* dkx = (const float*)d_in[1];
    const float* dky = (const float*)d_in[2];
    float* A  = (float*)d_out;
    float* ws = (float*)d_ws;

    // 1) dx scalar
    maxdx_kernel<<<1, 256, 0, stream>>>(c, ws);

    // 2) zero the whole matrix (out_size = 6912^2, divisible by 4)
    long long n4 = (long long)out_size / 4;
    wave_zero_kernel<<<8192, 256, 0, stream>>>(A, n4);

    // 3) scatter nonzeros (2304 lanes, <=11 stores each)
    scatter_kernel<<<(NN + 255) / 256, 256, 0, stream>>>(c, dkx, dky, ws, A);
}